// SimpleFuzzyAttention_3770981286265
// MI455X (gfx1250) — compile-verified
//
#include <hip/hip_runtime.h>
#include <hip/hip_bf16.h>

typedef __attribute__((ext_vector_type(2))) float v2f;
typedef __attribute__((ext_vector_type(4))) float v4f;
typedef __attribute__((ext_vector_type(8))) float v8f;

#define B_   32
#define L_   256
#define HID_ 4096

// ---------------------------------------------------------------------------
// K1: vbar[b][c] = (1/L) * sum_l value[b][l][c]      (32 x 4096 fp32)
// 131072 threads, one (b,c) each; loads coalesce across c each iteration.
// ---------------------------------------------------------------------------
__global__ __launch_bounds__(256) void k_reduce_value(const float* __restrict__ V,
                                                      float* __restrict__ vbar) {
    const int t = blockIdx.x * 256 + threadIdx.x;   // 0 .. 131071
    const int b = t >> 12;                          // / 4096
    const int c = t & 4095;
    const float* p = V + (size_t)b * L_ * HID_ + c;
    float s = 0.0f;
    #pragma unroll 8
    for (int l = 0; l < L_; ++l) s += p[(size_t)l * HID_];
    vbar[t] = s * (1.0f / (float)L_);
}

// ---------------------------------------------------------------------------
// K2/K3: Y[32][4096] = X[32][4096] @ W^T + bias, via V_WMMA_F32_16X16X4_F32.
// One wave per 16x16 tile: 2 M-tiles x 256 N-tiles = 512 waves = 64 blocks x 8.
// A (16x4 f32, v2f): lane m = lane%16; VGPR0/1 = K = k0 + 2*(lane>=16) + {0,1}.
// B (4x16  f32, v2f): B[k][n] = W[n][k]; lane n = lane%16, same K split.
// C/D (v8f): row = 16*m_tile + 8*(lane>=16) + r, col = 16*n_tile + lane%16.
// EXEC is all-1s: full waves, no divergence before the WMMA loop.
// ---------------------------------------------------------------------------
__global__ __launch_bounds__(256) void k_gemm_xwt(const float* __restrict__ X,
                                                  const float* __restrict__ W,
                                                  const float* __restrict__ bias,
                                                  float* __restrict__ Y) {
    const int K    = HID_;
    const int lane = threadIdx.x & 31;
    const int wave = threadIdx.x >> 5;
    const int tile = blockIdx.x * 8 + wave;     // 0 .. 511
    const int n_tile = tile & 255;
    const int m_tile = tile >> 8;               // 0 or 1
    const int ml = lane & 15;
    const int hi = lane >> 4;                   // lane-half selects K pair
    const float* xrow = X + (size_t)(m_tile * 16 + ml) * K + 2 * hi;
    const float* wrow = W + (size_t)(n_tile * 16 + ml) * K + 2 * hi;

    v8f acc = {};
    #pragma unroll 8
    for (int k0 = 0; k0 < K; k0 += 4) {
        v2f a = *(const v2f*)(xrow + k0);
        v2f b = *(const v2f*)(wrow + k0);
        // 8 args: (neg_a, A, neg_b, B, c_mod, C, reuse_a, reuse_b)
        acc = __builtin_amdgcn_wmma_f32_16x16x4_f32(
                  false, a, false, b, (short)0, acc, false, false);
    }

    const int n = n_tile * 16 + ml;
    const float bn = bias[n];
    #pragma unroll
    for (int r = 0; r < 8; ++r) {
        const int row = m_tile * 16 + hi * 8 + r;
        Y[(size_t)row * K + n] = acc[r] + bn;
    }
}

// ---------------------------------------------------------------------------
// K4: out[b][l][:] = o[b][:] (broadcast over l), attn[...] = 1/256 exactly.
// 268 MB of write-once data -> nontemporal b128 stores. o (512 KB) lives in L2.
// ---------------------------------------------------------------------------
__global__ __launch_bounds__(256) void k_writeout(const float* __restrict__ o,
                                                  float* __restrict__ out) {
    const unsigned t = blockIdx.x * 256u + threadIdx.x;   // 0 .. 16777215
    const unsigned HALF4 = 8388608u;                      // float4s in 'out' region
    if (t < HALF4) {
        const unsigned idx = t * 4u;                      // element index in out
        const unsigned b = idx >> 20;                     // / (L*HID) = 2^20
        const unsigned c = idx & (HID_ - 1);
        v4f v = *(const v4f*)(o + (size_t)b * HID_ + c);
        __builtin_nontemporal_store(v, (v4f*)(out + idx));
    } else {
        const unsigned idx = (t - HALF4) * 4u;
        const v4f v = {0.00390625f, 0.00390625f, 0.00390625f, 0.00390625f};
        __builtin_nontemporal_store(v, (v4f*)(out + 33554432u + idx));
    }
}

// ---------------------------------------------------------------------------
// inputs (setup_inputs order): 0 query, 1 key, 2 value, 3 Wq, 4 bq, 5 Wk,
// 6 bk, 7 Wv, 8 bv, 9 Wo, 10 bo, 11 centers, 12 widths
// d_out: out [32,256,4096] fp32 then attn [32,16,256,256] fp32 (== 1/256).
// ---------------------------------------------------------------------------
extern "C" void kernel_launch(void* const* d_in, const int* in_sizes, int n_in,
                              void* d_out, int out_size, void* d_ws, size_t ws_size,
                              hipStream_t stream) {
    const float* value = (const float*)d_in[2];
    const float* Wv    = (const float*)d_in[7];
    const float* bv    = (const float*)d_in[8];
    const float* Wo    = (const float*)d_in[9];
    const float* bo    = (const float*)d_in[10];
    float* out = (float*)d_out;

    float* vbar = (float*)d_ws;          // 32*4096 f32 = 512 KB
    float* tbuf = vbar + B_ * HID_;      // 512 KB
    float* obuf = tbuf + B_ * HID_;      // 512 KB

    k_reduce_value<<<512, 256, 0, stream>>>(value, vbar);
    k_gemm_xwt<<<64, 256, 0, stream>>>(vbar, Wv, bv, tbuf);
    k_gemm_xwt<<<64, 256, 0, stream>>>(tbuf, Wo, bo, obuf);
    k_writeout<<<65536, 256, 0, stream>>>(obuf, out);
}